// CapsuleLayer_30889404793494
// MI455X (gfx1250) — compile-verified
//
#include <hip/hip_runtime.h>
#include <hip/hip_bf16.h>

typedef float    v2f __attribute__((ext_vector_type(2)));
typedef float    v8f __attribute__((ext_vector_type(8)));
typedef _Float16 h8  __attribute__((ext_vector_type(8)));

#define B_    8
#define NIN   32
#define H_    14
#define CIN   8
#define CCAPS 32
#define COUT  16
#define RR    288          // NIN * 3 * 3
#define WW    12
#define SITES (B_*WW*WW)   // 1152
#define TILES (SITES/16)   // 72
#define NITER 3

__launch_bounds__(256)
__global__ void caps_routing_kernel(const float* __restrict__ x,
                                    const float* __restrict__ wt,
                                    float* __restrict__ out)
{
    // priors: [site(16)][r(288)][o(16)] as f16  -> 147456 B
    __shared__ _Float16 Plds[16 * RR * COUT];
    // reduction scratch: [site][j][o] f32       -> 16384 B   (total 160 KB)
    __shared__ float sred[16][16][16];

    const int c        = blockIdx.y;
    const int siteBase = blockIdx.x * 16;
    const int t    = threadIdx.x;
    const int lane = t & 31;
    const int wv   = t >> 5;      // wave 0..7
    const int m    = lane & 15;   // M row (site) for A, N col (o) for B
    const int hi   = lane >> 4;   // lane-half selects K pairs

    // ---------------- Stage 1: priors tiles via V_WMMA_F32_16X16X4_F32 ----
    {
        const int sg = siteBase + m;
        const int b  = sg / 144;
        const int pq = sg - b * 144;
        const int p  = pq / WW;
        const int q  = pq - p * WW;
        const float* xsite = x + b * (NIN*H_*H_*CIN) + p * (H_*CIN) + q * CIN;
        const float* wc    = wt + c * (RR*CIN*COUT);

        for (int rr = 0; rr < RR/8; ++rr) {
            const int r   = wv * (RR/8) + rr;
            const int n   = r / 9;
            const int rem = r - n*9;
            const int ky  = rem / 3;
            const int kx  = rem - ky*3;
            const float* xa = xsite + n*(H_*H_*CIN) + ky*(H_*CIN) + kx*CIN;
            const float* wr = wc + r*(CIN*COUT);

            // A 16x4 layout: lanes0-15 VGPR0=K0,VGPR1=K1 ; lanes16-31 K2,K3
            v2f a0, a1, b0, b1;
            a0.x = xa[2*hi + 0];  a0.y = xa[2*hi + 1];
            a1.x = xa[2*hi + 4];  a1.y = xa[2*hi + 5];
            // B 4x16: row K striped across lanes (N = m), mirrored halves
            b0.x = wr[(2*hi + 0)*COUT + m];  b0.y = wr[(2*hi + 1)*COUT + m];
            b1.x = wr[(2*hi + 4)*COUT + m];  b1.y = wr[(2*hi + 5)*COUT + m];

            v8f acc = {0.f,0.f,0.f,0.f,0.f,0.f,0.f,0.f};
            acc = __builtin_amdgcn_wmma_f32_16x16x4_f32(false, a0, false, b0,
                                                        (short)0, acc, false, false);
            acc = __builtin_amdgcn_wmma_f32_16x16x4_f32(false, a1, false, b1,
                                                        (short)0, acc, false, false);

            // D layout: VGPR j -> M=j (lanes0-15) / M=j+8 (lanes16-31), N=lane%16
            #pragma unroll
            for (int jj = 0; jj < 8; ++jj) {
                const int row = jj + 8*hi;
                Plds[(row*RR + r)*COUT + m] = (_Float16)acc[jj];
            }
        }
    }
    __syncthreads();

    // ---------------- Stage 2: dynamic routing (3 iters) ------------------
    const int site = t >> 4;   // 0..15 (2 sites per wave; 16-lane groups)
    const int j    = t & 15;   // lane-in-site
    const int sg   = siteBase + site;
    const int ob   = sg / 144;
    const int opq  = sg - ob * 144;
    float* outp = out + ((ob*CCAPS + c)*144 + opq) * COUT;

    float L[18];
    #pragma unroll
    for (int k = 0; k < 18; ++k) L[k] = 0.f;

    for (int it = 0; it < NITER; ++it) {
        // softmax over r for this site (logits identical across o)
        float mx = L[0];
        #pragma unroll
        for (int k = 1; k < 18; ++k) mx = fmaxf(mx, L[k]);
        #pragma unroll
        for (int msk = 1; msk < 16; msk <<= 1) mx = fmaxf(mx, __shfl_xor(mx, msk, 16));

        float e[18], z = 0.f;
        #pragma unroll
        for (int k = 0; k < 18; ++k) { e[k] = __expf(L[k] - mx); z += e[k]; }
        #pragma unroll
        for (int msk = 1; msk < 16; msk <<= 1) z += __shfl_xor(z, msk, 16);
        const float zinv = 1.f / z;

        // s[o] = sum_r prob[r] * P[r][o]   (partial over this thread's 18 r's)
        float s[16];
        #pragma unroll
        for (int oo = 0; oo < 16; ++oo) s[oo] = 0.f;
        #pragma unroll
        for (int k = 0; k < 18; ++k) {
            const int r = k*16 + j;
            const h8* prow = (const h8*)&Plds[(site*RR + r)*COUT];
            const h8 p0 = prow[0], p1 = prow[1];
            const float pr = e[k] * zinv;
            #pragma unroll
            for (int oo = 0; oo < 8; ++oo) {
                s[oo]   += pr * (float)p0[oo];
                s[oo+8] += pr * (float)p1[oo];
            }
        }
        #pragma unroll
        for (int oo = 0; oo < 16; ++oo) sred[site][j][oo] = s[oo];
        __syncthreads();

        // lane j gathers total s for o == j, then squash
        float sj = 0.f;
        #pragma unroll
        for (int kk = 0; kk < 16; ++kk) sj += sred[site][kk][j];
        float sn = sj * sj;
        #pragma unroll
        for (int msk = 1; msk < 16; msk <<= 1) sn += __shfl_xor(sn, msk, 16);
        const float vj = sj * (sn / (1.f + sn)) * rsqrtf(sn);

        if (it == NITER - 1) {
            outp[j] = vj;
        } else {
            __syncthreads();                 // step-5 reads done before reuse
            sred[site][0][j] = vj;           // broadcast v through scratch
            __syncthreads();
            float va[16];
            #pragma unroll
            for (int oo = 0; oo < 16; ++oo) va[oo] = sred[site][0][oo];
            __syncthreads();                 // reads done before next-iter writes

            #pragma unroll
            for (int k = 0; k < 18; ++k) {
                const int r = k*16 + j;
                const h8* prow = (const h8*)&Plds[(site*RR + r)*COUT];
                const h8 p0 = prow[0], p1 = prow[1];
                float d = 0.f;
                #pragma unroll
                for (int oo = 0; oo < 8; ++oo) {
                    d += va[oo]     * (float)p0[oo];
                    d += va[oo + 8] * (float)p1[oo];
                }
                L[k] += d;
            }
        }
    }
}

extern "C" void kernel_launch(void* const* d_in, const int* in_sizes, int n_in,
                              void* d_out, int out_size, void* d_ws, size_t ws_size,
                              hipStream_t stream) {
    const float* x  = (const float*)d_in[0];   // (8,32,14,14,8)
    const float* wt = (const float*)d_in[1];   // (32,288,8,16)
    float* out = (float*)d_out;                // (8,32,12,12,16)
    dim3 grid(TILES, CCAPS);                   // 72 site-tiles x 32 caps
    caps_routing_kernel<<<grid, 256, 0, stream>>>(x, wt, out);
}